// DecoupledTauNet_90340342104860
// MI455X (gfx1250) — compile-verified
//
#include <hip/hip_runtime.h>

// DecoupledTauNet fused kernel for gfx1250 (MI455X, wave32).
// f32 end-to-end; 128x128 matmuls on V_WMMA_F32_16X16X4_F32.
// Weights staged global->LDS via the Tensor Data Mover (tensor_load_to_lds)
// with hardware padding (128 dwords + 4 pad) reproducing the bank-conflict-free
// STR=132 layout. 256 threads / 128 points doubles resident waves for the
// L2-gather-bound trilinear sampling phase.

#define TPB   256           // threads per block (8 wave32)
#define TP    128           // points per block (2 threads per point)
#define HID   128
#define STR   132           // LDS row stride (floats): 128+4 -> conflict-free frags
#define NLEV  4
#define USE_TDM 1

typedef float v2f __attribute__((ext_vector_type(2)));
typedef float v8f __attribute__((ext_vector_type(8)));
typedef unsigned int u32x4 __attribute__((ext_vector_type(4)));
typedef int i32x8 __attribute__((ext_vector_type(8)));

__device__ __forceinline__ float silu_f(float x) { return x / (1.0f + expf(-x)); }

// Trilinear sample of one (16,S,S,S) grid; SiLU applied; writes dst[0..15] (LDS).
__device__ __forceinline__ void grid_feats_silu(const float* __restrict__ g, int S,
                                                float cx, float cy, float cz,
                                                float* __restrict__ dst) {
    float Sm1 = (float)(S - 1);
    float px = fminf(fmaxf((cx + 1.0f) * 0.5f * Sm1, 0.0f), Sm1);
    float py = fminf(fmaxf((cy + 1.0f) * 0.5f * Sm1, 0.0f), Sm1);
    float pz = fminf(fmaxf((cz + 1.0f) * 0.5f * Sm1, 0.0f), Sm1);
    int x0 = (int)px, y0 = (int)py, z0 = (int)pz;   // nonneg -> trunc == floor
    float wx = px - (float)x0, wy = py - (float)y0, wz = pz - (float)z0;
    int x1 = min(x0 + 1, S - 1), y1 = min(y0 + 1, S - 1), z1 = min(z0 + 1, S - 1);
    int i000 = (z0 * S + y0) * S + x0;
    int i001 = (z0 * S + y0) * S + x1;
    int i010 = (z0 * S + y1) * S + x0;
    int i011 = (z0 * S + y1) * S + x1;
    int i100 = (z1 * S + y0) * S + x0;
    int i101 = (z1 * S + y0) * S + x1;
    int i110 = (z1 * S + y1) * S + x0;
    int i111 = (z1 * S + y1) * S + x1;
    int S3 = S * S * S;
    #pragma unroll 4
    for (int c = 0; c < 16; ++c) {
        const float* gc = g + c * S3;
        float a000 = gc[i000], a001 = gc[i001], a010 = gc[i010], a011 = gc[i011];
        float a100 = gc[i100], a101 = gc[i101], a110 = gc[i110], a111 = gc[i111];
        float c00 = a000 + wx * (a001 - a000);
        float c01 = a010 + wx * (a011 - a010);
        float c10 = a100 + wx * (a101 - a100);
        float c11 = a110 + wx * (a111 - a110);
        float c0  = c00 + wy * (c01 - c00);
        float c1  = c10 + wy * (c11 - c10);
        float v   = c0 + wz * (c1 - c0);
        dst[c] = silu_f(v);
    }
}

// Fourier features for j in [j0, j0+cnt): dst[j] = sin(c . pe[j] + bias[j])
__device__ __forceinline__ void ffm_sin(const float* __restrict__ pe,
                                        const float* __restrict__ bias,
                                        float cx, float cy, float cz,
                                        float* __restrict__ dst, int j0, int cnt) {
    #pragma unroll 4
    for (int j = j0; j < j0 + cnt; ++j) {
        float s = cx * pe[j * 3 + 0] + cy * pe[j * 3 + 1] + cz * pe[j * 3 + 2];
        if (bias) s += bias[j];
        dst[j] = sinf(s);
    }
}

// ---- Tensor Data Mover: stage 128x128 f32 weight matrix global->LDS with
// hardware row padding (128 dwords data + 4 dwords pad == STR 132). ----
__device__ __forceinline__ void tdm_issue(const float* __restrict__ Wg,
                                          float* WsLds, int t) {
#if USE_TDM
    if (t < 32) {                       // wave 0 only; TDM ignores EXEC (1 issue/wave)
        unsigned int lds = (unsigned int)(unsigned long long)(uintptr_t)WsLds;
        unsigned long long ga = (unsigned long long)(uintptr_t)Wg;
        u32x4 g0;
        g0[0] = 1u;                                   // count=1 (valid user D#)
        g0[1] = lds;                                  // lds_addr (bytes)
        g0[2] = (unsigned int)ga;                     // global_addr[31:0]
        g0[3] = ((unsigned int)(ga >> 32) & 0x01FFFFFFu) | (2u << 30); // addr[56:32]|type=2
        i32x8 g1;
        g1[0] = (2 << 16)    // data_size = 4 bytes
              | (1 << 20)    // pad_enable
              | (6 << 22)    // pad_interval: 128 dwords
              | (3 << 25);   // pad_amount: 4 dwords
        g1[1] = 128 << 16;   // tensor_dim0 = 128 (bits 79:48 -> low 16 here)
        g1[2] = 128 << 16;   // tensor_dim0 hi = 0 | tensor_dim1 = 128 (bits 111:80)
        g1[3] = 128 << 16;   // tensor_dim1 hi = 0 | tile_dim0 = 128 (bits 127:112)
        g1[4] = 128;         // tile_dim1 = 128, tile_dim2 = 0
        g1[5] = 128;         // tensor_dim0_stride = 128 (48b, bits 207:160)
        g1[6] = 0;           // stride0 hi | tensor_dim1_stride lo (unused for 2D)
        g1[7] = 0;
        asm volatile("tensor_load_to_lds %0, %1" :: "s"(g0), "s"(g1) : "memory");
    }
#else
    #pragma unroll 4
    for (int i = t; i < HID * HID; i += TPB) {
        int nr = i >> 7, k = i & 127;
        WsLds[nr * STR + k] = Wg[i];
    }
#endif
}

__device__ __forceinline__ void tdm_wait(int t) {
#if USE_TDM
    if (t < 32) __builtin_amdgcn_s_wait_tensorcnt(0);
#endif
}

// Y[TP x 128] = A[TP x 128] @ W.T via V_WMMA_F32_16X16X4_F32 (all LDS operands)
__device__ __forceinline__ void gemm128(const float* __restrict__ A,
                                        const float* __restrict__ W,
                                        float* __restrict__ Y, int t) {
    int lane = t & 31;
    int wave = t >> 5;
    int lm = lane & 15;
    int lh = lane >> 4;          // selects K pair (k0,k0+1) vs (k0+2,k0+3)
    for (int rt = wave; rt < TP / 16; rt += TPB / 32) {
        int m0 = rt * 16;
        const float* Arow = A + (m0 + lm) * STR + 2 * lh;
        for (int nt = 0; nt < HID / 16; ++nt) {
            int n0 = nt * 16;
            const float* Brow = W + (n0 + lm) * STR + 2 * lh;  // B[k][n] = W[n][k]
            v8f acc = {0.f, 0.f, 0.f, 0.f, 0.f, 0.f, 0.f, 0.f};
            #pragma unroll
            for (int k0 = 0; k0 < HID; k0 += 4) {
                v2f a = *(const v2f*)(Arow + k0);
                v2f b = *(const v2f*)(Brow + k0);
                acc = __builtin_amdgcn_wmma_f32_16x16x4_f32(
                    false, a, false, b, (short)0, acc, false, false);
            }
            float* Yp = Y + (m0 + 8 * lh) * STR + n0 + lm;  // D: vgpr r -> row m0+8*lh+r
            #pragma unroll
            for (int r = 0; r < 8; ++r) Yp[r * STR] = acc[r];
        }
    }
}

__global__ __launch_bounds__(TPB, 1)
void dtn_fused_kernel(const float* __restrict__ phys,
                      const float* __restrict__ net,
                      const float* __restrict__ tau_in,
                      const float* __restrict__ vg0, const float* __restrict__ vg1,
                      const float* __restrict__ vg2, const float* __restrict__ vg3,
                      const float* __restrict__ tg0, const float* __restrict__ tg1,
                      const float* __restrict__ tg2, const float* __restrict__ tg3,
                      const float* __restrict__ vec_pe_w,
                      const float* __restrict__ tau_pe_w,
                      const float* __restrict__ tau_pe_b,
                      const float* __restrict__ vec_scale,
                      const float* __restrict__ vec_map_w,
                      const float* __restrict__ tau_mlp_w,
                      const float* __restrict__ vec_mlp_w,
                      const float* __restrict__ tau_params,
                      const float* __restrict__ last_w,
                      float* __restrict__ out, int n) {
    __shared__ float Xs[TP * STR];
    __shared__ float Ts[TP * STR];
    __shared__ float Ys[TP * STR];
    __shared__ float Ws[HID * STR];

    int t   = threadIdx.x;
    int pid = t >> 1;            // point within block (2 threads / point)
    int half = t & 1;            // which half of the 128 features/levels
    int j0  = half * 64;         // elementwise column range [j0, j0+64)
    int p = blockIdx.x * TP + pid;
    bool valid = p < n;
    int pc = valid ? p : (n - 1);

    float cx = net[pc * 3 + 0], cy = net[pc * 3 + 1], cz = net[pc * 3 + 2];
    float tau = tau_in[pc];

    const float* vgrids[NLEV] = {vg0, vg1, vg2, vg3};
    const float* tgrids[NLEV] = {tg0, tg1, tg2, tg3};
    const int SZ[NLEV] = {16, 32, 64, 96};

    // Warm L2 for the next weight matrix while we gather (global_prefetch_b8).
    for (int i = t * 32; i < HID * HID; i += TPB * 32)
        __builtin_prefetch(vec_map_w + i, 0, 1);

    // Stage tau_mlp_w via TDM; DMA overlaps the tau gather below.
    tdm_issue(tau_mlp_w, Ws, t);

    // tau branch features -> Ts (each thread: 2 grid levels + half the FFM)
    for (int l = 2 * half; l < 2 * half + 2; ++l)
        grid_feats_silu(tgrids[l], SZ[l], cx, cy, cz, &Ts[pid * STR + l * 16]);
    ffm_sin(tau_pe_w, tau_pe_b, cx, cy, cz, &Ts[pid * STR + 64], half * 32, 32);

    tdm_wait(t);
    __syncthreads();
    gemm128(Ts, Ws, Ys, t);              // tmul = T @ tau_mlp_w.T
    __syncthreads();

    tdm_issue(vec_map_w, Ws, t);         // Ws free again after barrier
    for (int j = j0; j < j0 + 64; ++j) Ts[pid * STR + j] = Ys[pid * STR + j];
    // vec branch features -> Xs
    for (int l = 2 * half; l < 2 * half + 2; ++l)
        grid_feats_silu(vgrids[l], SZ[l], cx, cy, cz, &Xs[pid * STR + l * 16]);
    ffm_sin(vec_pe_w, nullptr, cx, cy, cz, &Xs[pid * STR + 64], half * 32, 32);
    tdm_wait(t);
    __syncthreads();
    gemm128(Xs, Ws, Ys, t);              // vec_map layer 0
    __syncthreads();

    tdm_issue(vec_map_w + HID * HID, Ws, t);
    for (int j = j0; j < j0 + 64; ++j) Xs[pid * STR + j] = silu_f(Ys[pid * STR + j]);
    tdm_wait(t);
    __syncthreads();
    gemm128(Xs, Ws, Ys, t);              // vec_map layer 1
    __syncthreads();

    tdm_issue(vec_mlp_w, Ws, t);
    // out = tanh(tau * vec_scale * silu(h2))
    for (int j = j0; j < j0 + 64; ++j)
        Xs[pid * STR + j] = tanhf(tau * vec_scale[j] * silu_f(Ys[pid * STR + j]));
    tdm_wait(t);
    __syncthreads();
    gemm128(Xs, Ws, Ys, t);              // vec_mlp layer 0
    __syncthreads();

    tdm_issue(vec_mlp_w + HID * HID, Ws, t);
    for (int j = j0; j < j0 + 64; ++j) {
        float h = Ys[pid * STR + j] * Ts[pid * STR + j];   // gated by tau features
        float tt = tanhf(tau * tau_params[0 * HID + j]);
        Xs[pid * STR + j] += tt * silu_f(h);
    }
    tdm_wait(t);
    __syncthreads();
    gemm128(Xs, Ws, Ys, t);              // vec_mlp layer 1
    __syncthreads();

    tdm_issue(vec_mlp_w + 2 * HID * HID, Ws, t);
    for (int j = j0; j < j0 + 64; ++j) {
        float tt = tanhf(tau * tau_params[1 * HID + j]);
        Xs[pid * STR + j] += tt * silu_f(Ys[pid * STR + j]);
    }
    tdm_wait(t);
    __syncthreads();
    gemm128(Xs, Ws, Ys, t);              // vec_mlp layer 2
    __syncthreads();

    for (int j = j0; j < j0 + 64; ++j) {
        float tt = tanhf(tau * tau_params[2 * HID + j]);
        Xs[pid * STR + j] += tt * silu_f(Ys[pid * STR + j]);
    }

    // final: out = phys + X @ last_w.T ; pair-reduce partials with wave shuffle
    float o0 = 0.f, o1 = 0.f, o2 = 0.f;
    for (int j = j0; j < j0 + 64; ++j) {
        float xv = Xs[pid * STR + j];
        o0 += xv * last_w[0 * HID + j];
        o1 += xv * last_w[1 * HID + j];
        o2 += xv * last_w[2 * HID + j];
    }
    o0 += __shfl_xor(o0, 1, 32);
    o1 += __shfl_xor(o1, 1, 32);
    o2 += __shfl_xor(o2, 1, 32);
    if (valid && half == 0) {
        out[p * 3 + 0] = phys[p * 3 + 0] + o0;
        out[p * 3 + 1] = phys[p * 3 + 1] + o1;
        out[p * 3 + 2] = phys[p * 3 + 2] + o2;
    }
}

extern "C" void kernel_launch(void* const* d_in, const int* in_sizes, int n_in,
                              void* d_out, int out_size, void* d_ws, size_t ws_size,
                              hipStream_t stream) {
    const float* phys       = (const float*)d_in[0];
    const float* net        = (const float*)d_in[1];
    const float* tau        = (const float*)d_in[2];
    const float* vg0        = (const float*)d_in[3];
    const float* vg1        = (const float*)d_in[4];
    const float* vg2        = (const float*)d_in[5];
    const float* vg3        = (const float*)d_in[6];
    const float* tg0        = (const float*)d_in[7];
    const float* tg1        = (const float*)d_in[8];
    const float* tg2        = (const float*)d_in[9];
    const float* tg3        = (const float*)d_in[10];
    const float* vec_pe_w   = (const float*)d_in[11];
    const float* tau_pe_w   = (const float*)d_in[12];
    const float* tau_pe_b   = (const float*)d_in[13];
    const float* vec_scale  = (const float*)d_in[14];
    const float* vec_map_w  = (const float*)d_in[15];
    const float* tau_mlp_w  = (const float*)d_in[16];
    const float* vec_mlp_w  = (const float*)d_in[17];
    const float* tau_params = (const float*)d_in[18];
    const float* last_w     = (const float*)d_in[19];
    float* out = (float*)d_out;

    int n = in_sizes[1] / 3;
    int grid = (n + TP - 1) / TP;
    dtn_fused_kernel<<<dim3(grid), dim3(TPB), 0, stream>>>(
        phys, net, tau, vg0, vg1, vg2, vg3, tg0, tg1, tg2, tg3,
        vec_pe_w, tau_pe_w, tau_pe_b, vec_scale, vec_map_w, tau_mlp_w,
        vec_mlp_w, tau_params, last_w, out, n);
}